// Supersampling_41128606827041
// MI455X (gfx1250) — compile-verified
//
#include <hip/hip_runtime.h>

// Problem constants (fixed by the reference).
#define FEATC 6
#define OUTC  3
#define KS    3
#define IMH   512
#define IMW   512
#define HW    (IMH * IMW)
#define PATCH (FEATC * KS * KS)   // 54

// Tile: 32x8 pixels per 256-thread block (8 wave32s, one wave per pixel row).
#define TW      32
#define TH      8
#define HALO_W  (TW + 2)          // 34
#define HALO_H  (TH + 2)          // 10
#define PLANE   (HALO_W * HALO_H) // 340
#define LDS_ELEMS (FEATC * PLANE) // 2040 floats = 8160 B

typedef __attribute__((address_space(1))) int global_i32;
typedef __attribute__((address_space(3))) int lds_i32;

__device__ __forceinline__ float nt_load(const float* p) {
  // kernel_map is streamed once (680 MB >> 192 MB L2): non-temporal so it
  // doesn't evict the cache-resident feat (25 MB, reused 9x).
  return __builtin_nontemporal_load(p);
}

__device__ __forceinline__ void feat_to_lds(const float* gp, float* lp) {
#if defined(__has_builtin) && __has_builtin(__builtin_amdgcn_global_load_async_to_lds_b32)
  // CDNA5 async copy path: global -> LDS without VGPR round trip (ASYNCcnt).
  __builtin_amdgcn_global_load_async_to_lds_b32(
      (global_i32*)gp, (lds_i32*)lp, /*imm offset*/ 0, /*cpol*/ 0);
#else
  *lp = *gp;  // fallback: load + ds_store
#endif
}

__device__ __forceinline__ void wait_async0() {
#if defined(__has_builtin) && __has_builtin(__builtin_amdgcn_s_wait_asynccnt)
  __builtin_amdgcn_s_wait_asynccnt(0);
#else
  asm volatile("s_wait_asynccnt 0" ::: "memory");
#endif
}

__global__ __launch_bounds__(256) void kpn_filter_kernel(
    const float* __restrict__ feat,   // [B, 6, 512, 512]
    const float* __restrict__ km,     // [B, 162, 512, 512]
    float* __restrict__ out) {        // [B, 3, 512, 512]
  __shared__ float s_feat[LDS_ELEMS];

  const int tx = threadIdx.x;          // 0..31 (fast along W -> coalescing)
  const int ty = threadIdx.y;          // 0..7
  const int b  = blockIdx.z;
  const int w0 = blockIdx.x * TW;
  const int h0 = blockIdx.y * TH;
  const int tid = ty * TW + tx;

  const float* featb = feat + (size_t)b * FEATC * HW;

  // Cooperative fill of the 6 x 10 x 34 halo tile. OOB (zero-pad) slots get
  // ds_store zeros; interior slots get async global->LDS loads. The two sets
  // of LDS addresses are disjoint, so ordering between them doesn't matter;
  // both are fenced by s_wait_asynccnt + the workgroup barrier below.
  for (int i = tid; i < LDS_ELEMS; i += 256) {
    const int c   = i / PLANE;
    const int rem = i - c * PLANE;
    const int y   = rem / HALO_W;
    const int x   = rem - y * HALO_W;
    const int gy  = h0 + y - 1;
    const int gx  = w0 + x - 1;
    if ((unsigned)gy < (unsigned)IMH && (unsigned)gx < (unsigned)IMW) {
      feat_to_lds(featb + ((size_t)c * IMH + gy) * IMW + gx, &s_feat[i]);
    } else {
      s_feat[i] = 0.0f;
    }
  }
  wait_async0();
  __syncthreads();

  const int h = h0 + ty;
  const int w = w0 + tx;
  // Per-pixel base into kernel_map; plane k lives at kmb + k*HW.
  const float* kmb = km + (size_t)b * (OUTC * PATCH) * HW + (size_t)h * IMW + w;

  float a0 = 0.0f, a1 = 0.0f, a2 = 0.0f;
#pragma unroll
  for (int c = 0; c < FEATC; ++c) {
#pragma unroll
    for (int ky = 0; ky < KS; ++ky) {
#pragma unroll
      for (int kx = 0; kx < KS; ++kx) {
        // Lanes (consecutive tx) read consecutive LDS dwords: conflict-free.
        const float p = s_feat[c * PLANE + (ty + ky) * HALO_W + (tx + kx)];
        const int k = c * (KS * KS) + ky * KS + kx;
        a0 = fmaf(p, nt_load(kmb + (size_t)k * HW), a0);
        a1 = fmaf(p, nt_load(kmb + (size_t)(k + PATCH) * HW), a1);
        a2 = fmaf(p, nt_load(kmb + (size_t)(k + 2 * PATCH) * HW), a2);
      }
    }
  }

  float* ob = out + (size_t)b * OUTC * HW + (size_t)h * IMW + w;
  __builtin_nontemporal_store(a0, ob);
  __builtin_nontemporal_store(a1, ob + (size_t)HW);
  __builtin_nontemporal_store(a2, ob + (size_t)(2 * HW));
}

extern "C" void kernel_launch(void* const* d_in, const int* in_sizes, int n_in,
                              void* d_out, int out_size, void* d_ws, size_t ws_size,
                              hipStream_t stream) {
  const float* feat = (const float*)d_in[0];
  const float* km   = (const float*)d_in[1];
  float* out        = (float*)d_out;

  const int B = in_sizes[0] / (FEATC * HW);  // = 4 for the reference shapes

  dim3 block(TW, TH, 1);                       // 256 threads = 8 wave32s
  dim3 grid(IMW / TW, IMH / TH, B);            // 16 x 64 x 4 blocks
  kpn_filter_kernel<<<grid, block, 0, stream>>>(feat, km, out);
}